// QKV_Adapter_ViT_audio_filter_58909771432050
// MI455X (gfx1250) — compile-verified
//
#include <hip/hip_runtime.h>
#include <math.h>

typedef __attribute__((ext_vector_type(2))) float v2f;
typedef __attribute__((ext_vector_type(8))) float v8f;

#define DD    768      // token dim
#define DH    384      // reduced dim
#define NPOS  20       // spatial positions (4x5)
#define PS    33       // padded position stride in LDS (bank spread)
#define T1S   42       // 6*7 zero-padded image per channel
#define WW    5

__device__ __forceinline__ float gelu_exact(float x) {
    return 0.5f * x * (1.0f + erff(x * 0.70710678118654752440f));
}

__device__ __forceinline__ v8f wmma4(v2f a, v2f b, v8f c) {
    // V_WMMA_F32_16X16X4_F32: D = A(16x4 f32) * B(4x16 f32) + C
    return __builtin_amdgcn_wmma_f32_16x16x4_f32(
        /*neg_a=*/false, a, /*neg_b=*/false, b,
        /*c_mod=*/(short)0, c, /*reuse_a=*/false, /*reuse_b=*/false);
}

// Build the 20x20 complex linear operator equivalent to
// ifftn(roll(fftn(t),(2,2)) * mask) applied over the (4,5) spatial grid.
// Mr at fmat[0..399], Mi at fmat[400..799].
__global__ void filter_matrix_kernel(const float* __restrict__ mask,
                                     float* __restrict__ fmat) {
    int t = blockIdx.x * blockDim.x + threadIdx.x;
    if (t >= 400) return;
    int pp = t / NPOS, p = t % NPOS;
    int hp = pp / WW, wp = pp % WW;
    int h  = p  / WW, w  = p  % WW;
    float re = 0.f, im = 0.f;
    for (int a = 0; a < 4; ++a) {
        for (int bq = 0; bq < 5; ++bq) {
            float mv = mask[a * 5 + bq];
            if (mv != 0.f) {
                int ky = (a + 2) & 3;       // (a-2) mod 4
                int kx = (bq + 3) % 5;      // (b-2) mod 5
                float ang = 6.283185307179586f *
                    ((float)(a * hp) * 0.25f + (float)(bq * wp) * 0.2f
                   - (float)(ky * h) * 0.25f - (float)(kx * w) * 0.2f);
                re += mv * __cosf(ang) == 0.f ? mv * cosf(ang) : mv * cosf(ang);
                im += mv * sinf(ang);
            }
        }
    }
    fmat[t]       = re * 0.05f;   // includes 1/(H*W)
    fmat[400 + t] = im * 0.05f;
}

extern "C" __global__ void __launch_bounds__(256)
qkv_adapter_fused(const float* __restrict__ x,
                  const float* __restrict__ w1,
                  const float* __restrict__ w2,
                  const float* __restrict__ wq,
                  const float* __restrict__ wk,
                  const float* __restrict__ wv,
                  const float* __restrict__ fmat,
                  float* __restrict__ out,
                  int batch) {
    extern __shared__ float smem[];
    float* bufA  = smem;            // 768*33 floats: x cols, then t2/t3, stage-4 B
    float* bufT1 = smem + DD * PS;  // 384*42 floats: padded t1 image; later Mr/Mi

    const int b    = blockIdx.x;
    const int tid  = threadIdx.x;
    const int wave = tid >> 5;
    const int lane = tid & 31;
    const int lo   = lane & 15;
    const int hi   = lane >> 4;

    // zero both LDS buffers (borders of t1 image + position padding)
    for (int i = tid; i < DD * PS + DH * T1S; i += 256) smem[i] = 0.f;
    __syncthreads();

    // load x (skip CLS token) -> bufA[c*PS + p], coalesced on c
    const float* xb = x + (size_t)b * 21 * DD + DD;
    for (int i = tid; i < NPOS * DD; i += 256) {
        int p = i / DD, c = i % DD;
        bufA[c * PS + p] = xb[p * DD + c];
    }
    __syncthreads();

    // ---------- stage 1: t1 = gelu(w1 @ x)  (384x20), into padded image ----------
    for (int job = wave; job < 48; job += 8) {
        int tm = job >> 1, tn = job & 1;
        int m   = tm * 16 + lo;
        int col = tn * 16 + lo;
        const float* Arow = w1 + (size_t)m * DD;
        v8f acc = {0.f, 0.f, 0.f, 0.f, 0.f, 0.f, 0.f, 0.f};
#pragma unroll 4
        for (int k0 = 0; k0 < DD; k0 += 4) {
            int ka = k0 + 2 * hi;
            v2f a = *(const v2f*)(Arow + ka);
            v2f bb;
            bb.x = bufA[ka * PS + col];
            bb.y = bufA[(ka + 1) * PS + col];
            acc = wmma4(a, bb, acc);
        }
        if (col < NPOS) {
            int h = col / WW, w = col % WW;
            int off = (h + 1) * 7 + (w + 1);
#pragma unroll
            for (int j = 0; j < 8; ++j) {
                int o = tm * 16 + j + 8 * hi;
                bufT1[o * T1S + off] = gelu_exact(acc[j]);
            }
        }
    }
    __syncthreads();

    // ---------- stage 2: t2 = gelu(conv3x3(t1))  (768x20) -> bufA ----------
    for (int job = wave; job < 96; job += 8) {
        int tm = job >> 1, tn = job & 1;
        int m   = tm * 16 + lo;
        int col = tn * 16 + lo;
        int hh = col / WW, ww = col % WW;
        v8f acc = {0.f, 0.f, 0.f, 0.f, 0.f, 0.f, 0.f, 0.f};
        for (int dy = 0; dy < 3; ++dy) {
            for (int dx = 0; dx < 3; ++dx) {
                // invalid columns read the zero border at offset 0
                int boff = (col < NPOS) ? ((hh + dy) * 7 + (ww + dx)) : 0;
                const float* Aw = w2 + (size_t)m * DH * 9 + dy * 3 + dx;
#pragma unroll 4
                for (int k0 = 0; k0 < DH; k0 += 4) {
                    int ka = k0 + 2 * hi;
                    v2f a;
                    a.x = Aw[ka * 9];
                    a.y = Aw[(ka + 1) * 9];
                    v2f bb;
                    bb.x = bufT1[ka * T1S + boff];
                    bb.y = bufT1[(ka + 1) * T1S + boff];
                    acc = wmma4(a, bb, acc);
                }
            }
        }
#pragma unroll
        for (int j = 0; j < 8; ++j) {
            int o = tm * 16 + j + 8 * hi;
            bufA[o * PS + col] = gelu_exact(acc[j]);
        }
    }
    __syncthreads();

    // ---------- stage 3: FFT low-pass filter as 20x20 complex matrix, |.| ----------
    float* fM = bufT1;  // t1 image no longer needed; stage Mr/Mi in LDS
    for (int i = tid; i < 800; i += 256) fM[i] = fmat[i];
    __syncthreads();
    for (int c = tid; c < DD; c += 256) {
        float v[NPOS];
#pragma unroll
        for (int p = 0; p < NPOS; ++p) v[p] = bufA[c * PS + p];
        float o[NPOS];
#pragma unroll
        for (int pp = 0; pp < NPOS; ++pp) {
            float re = 0.f, im = 0.f;
#pragma unroll
            for (int p = 0; p < NPOS; ++p) {
                re += fM[pp * NPOS + p] * v[p];
                im += fM[400 + pp * NPOS + p] * v[p];
            }
            o[pp] = sqrtf(re * re + im * im);
        }
#pragma unroll
        for (int pp = 0; pp < NPOS; ++pp) bufA[c * PS + pp] = o[pp];
    }
    __syncthreads();

    // ---------- stage 4: q/k/v = W @ t3, scatter to (3,B,12,20,64) ----------
    const size_t obstride = (size_t)12 * NPOS * 64;  // 15360 per batch per gemm
    for (int job = wave; job < 288; job += 8) {
        int g = job / 96;
        int r = job % 96;
        int tm = r >> 1, tn = r & 1;
        const float* W = (g == 0) ? wq : (g == 1) ? wk : wv;
        int m   = tm * 16 + lo;
        int col = tn * 16 + lo;
        const float* Arow = W + (size_t)m * DD;
        v8f acc = {0.f, 0.f, 0.f, 0.f, 0.f, 0.f, 0.f, 0.f};
#pragma unroll 4
        for (int k0 = 0; k0 < DD; k0 += 4) {
            int ka = k0 + 2 * hi;
            v2f a = *(const v2f*)(Arow + ka);
            v2f bb;
            bb.x = bufA[ka * PS + col];
            bb.y = bufA[(ka + 1) * PS + col];
            acc = wmma4(a, bb, acc);
        }
        if (col < NPOS) {
            float* og = out + (size_t)g * batch * obstride + (size_t)b * obstride;
#pragma unroll
            for (int j = 0; j < 8; ++j) {
                int o = tm * 16 + j + 8 * hi;           // output channel
                og[(size_t)(o >> 6) * (NPOS * 64) + col * 64 + (o & 63)] = acc[j];
            }
        }
    }
}

extern "C" void kernel_launch(void* const* d_in, const int* in_sizes, int n_in,
                              void* d_out, int out_size, void* d_ws, size_t ws_size,
                              hipStream_t stream) {
    const float* x    = (const float*)d_in[0];
    const float* w1   = (const float*)d_in[1];
    const float* w2   = (const float*)d_in[2];
    const float* wq   = (const float*)d_in[3];
    const float* wk   = (const float*)d_in[4];
    const float* wv   = (const float*)d_in[5];
    const float* mask = (const float*)d_in[6];
    (void)n_in; (void)out_size; (void)ws_size;

    int batch = in_sizes[0] / (21 * DD);
    float* fmat = (float*)d_ws;  // Mr[400] ++ Mi[400]

    filter_matrix_kernel<<<2, 256, 0, stream>>>(mask, fmat);

    size_t lds_bytes = (size_t)(DD * PS + DH * T1S) * sizeof(float);  // ~162 KB of 320 KB WGP LDS
    (void)hipFuncSetAttribute((const void*)qkv_adapter_fused,
                              hipFuncAttributeMaxDynamicSharedMemorySize,
                              (int)lds_bytes);
    qkv_adapter_fused<<<batch, 256, lds_bytes, stream>>>(
        x, w1, w2, wq, wk, wv, fmat, (float*)d_out, batch);
}